// WeightedMinimumPowerDistortionlessResponseFilter_3977139716229
// MI455X (gfx1250) — compile-verified
//
#include <hip/hip_runtime.h>

// WMPDR beamformer for MI455X (gfx1250, wave32).
// Dominant work: 7 weighted-Gram passes (B*F batched (8xT)(Tx8) complex Gram),
// done with V_WMMA_F32_16X16X4_F32: [xr;xi] stacked as 16 rows, one WMMA per
// 4 time steps, fp32 accumulation.
// Parallelism: 4 waves per (b,f), each Gram-accumulating a T/4 slice (partials
// summed in LDS during the re/im recombination) -> 4112 waves total so L2
// latency is hidden by wave interleaving; plus cross-iteration double
// buffering so each batch of 10 b64 loads overlaps the previous batch's WMMAs.
// EXEC is all-ones at every WMMA (full waves, uniform control flow).

#define BB 4
#define CC 8
#define FF 257
#define TT 2000
#define BF (BB * FF)
#define NUM_ITER 5
#define DIAG_REG 1e-6f
#define EPS 1e-8f
#define PSD_EPS 1e-15f
#define GCH 5                        // chunks (of 4 timesteps) per load batch
#define TQ (TT / 4)                  // timesteps per wave (T split 4 ways)

typedef float v2f __attribute__((ext_vector_type(2)));
typedef float v8f __attribute__((ext_vector_type(8)));

// ---------------------------------------------------------------- mask sums
__global__ void msum_kernel(const float* __restrict__ mask, float* __restrict__ msum) {
    int bf = blockIdx.x * blockDim.x + threadIdx.x;
    if (bf >= BF) return;
    const float* mp = mask + (size_t)bf * TT;
    float s = 0.f;
    for (int t = 0; t < TT; ++t) s += mp[t];
    msum[bf] = s;
}

// ------------------------------------------------- weighted PSD via WMMA f32
// mode 0: w = mask[t]/(msum+PSD_EPS)        (signal PSD)
// mode 1: w = (1-mask[t])/(T-msum+PSD_EPS)  (noise PSD)
// mode 2: w = wsrc[t]/T                     (loop weighted PSD)
// One block per (b,f); wave q of the block accumulates t in [q*TQ,(q+1)*TQ).
__global__ __launch_bounds__(128) void psd_wmma_kernel(
    const float* __restrict__ xr, const float* __restrict__ xi,
    const float* __restrict__ wsrc, const float* __restrict__ msum,
    float* __restrict__ psd_out, int mode)
{
    const int lane = threadIdx.x & 31;
    const int wib  = threadIdx.x >> 5;           // wave in block = T-quarter
    const int bf   = blockIdx.x;
    const int b = bf / FF, f = bf - b * FF;

    float scale, wbase = 0.f, wsign = 1.f;
    if (mode == 0)      scale = 1.0f / (msum[bf] + PSD_EPS);
    else if (mode == 1) { scale = 1.0f / ((float)TT - msum[bf] + PSD_EPS); wbase = 1.f; wsign = -1.f; }
    else                scale = 1.0f / (float)TT;

    // A-matrix (16x4 f32) lane layout: lanes 0-15 -> K=0,1 ; lanes 16-31 -> K=2,3.
    // Row m = lane&15: rows 0..7 = xr channels, rows 8..15 = xi channels.
    const int m  = lane & 15;
    const int kb = (lane >> 4) << 1;             // 0 or 2
    const int ch = (m < 8) ? m : (m - 8);
    const float* base = (m < 8) ? xr : xi;
    const float* up = base + (((size_t)b * CC + ch) * FF + f) * (size_t)TT + (size_t)wib * TQ + kb;
    const float* wp = wsrc + ((size_t)b * FF + f) * (size_t)TT + (size_t)wib * TQ + kb;

    v8f acc = {0.f, 0.f, 0.f, 0.f, 0.f, 0.f, 0.f, 0.f};

    // 125 chunks per wave -> 25 batches of GCH=5 chunks, double buffered:
    // batch i+1's 10 b64 loads are issued while batch i's WMMAs execute.
    v2f abuf[GCH], wbuf[GCH];
    #pragma unroll
    for (int g = 0; g < GCH; ++g) {
        abuf[g] = *(const v2f*)(up + 4 * g);
        wbuf[g] = *(const v2f*)(wp + 4 * g);
    }
    for (int t0 = 0; t0 < TQ; t0 += 4 * GCH) {
        const int tn = t0 + 4 * GCH;
        const bool more = tn < TQ;
        v2f anx[GCH], wnx[GCH];
        if (more) {
            __builtin_prefetch(up + tn + 4 * GCH, 0, 0);   // global_prefetch
            #pragma unroll
            for (int g = 0; g < GCH; ++g) {
                anx[g] = *(const v2f*)(up + tn + 4 * g);
                wnx[g] = *(const v2f*)(wp + tn + 4 * g);
            }
        }
        #pragma unroll
        for (int g = 0; g < GCH; ++g) {
            v2f a = abuf[g];
            v2f bm;                              // B mirrors A layout: lane&15 = N
            bm.x = a.x * ((wbase + wsign * wbuf[g].x) * scale);
            bm.y = a.y * ((wbase + wsign * wbuf[g].y) * scale);
            acc = __builtin_amdgcn_wmma_f32_16x16x4_f32(
                false, a, false, bm, (short)0, acc, false, false);
        }
        if (more) {
            #pragma unroll
            for (int g = 0; g < GCH; ++g) { abuf[g] = anx[g]; wbuf[g] = wnx[g]; }
        }
    }

    // Partial D tiles: D[r + 8*(lane>>4)][lane&15] = acc[r] -> LDS
    __shared__ float Dls[4][16][16];
    const int rowoff = (lane >> 4) << 3;
    for (int r = 0; r < 8; ++r)
        Dls[wib][r + rowoff][lane & 15] = acc[r];
    __syncthreads();

    // Sum the 4 T-slice partials and recombine:
    // P_re[c][e] = D[c][e] + D[c+8][e+8] ; P_im[c][e] = D[c+8][e] - D[c][e+8]
    if (threadIdx.x < 64) {
        int idx = threadIdx.x;                   // 0..63
        int c = idx >> 3, e = idx & 7;
        float re = 0.f, im = 0.f;
        #pragma unroll
        for (int q = 0; q < 4; ++q) {
            re += Dls[q][c][e]     + Dls[q][c + 8][e + 8];
            im += Dls[q][c + 8][e] - Dls[q][c][e + 8];
        }
        size_t o = ((size_t)bf * 64 + idx) * 2;
        psd_out[o]     = re;
        psd_out[o + 1] = im;
    }
}

// ---------------------------------------------- steering vector (per (b,f))
// chol(psd_n)=L ; invL ; whiten psd_s ; power iteration -> principal eigvec ;
// sv = L v.   1028 independent 8x8 complex problems -> one thread each.
__global__ void steering_kernel(const float* __restrict__ psd_s_g,
                                const float* __restrict__ psd_n_g,
                                float* __restrict__ sv_g)
{
    int bf = blockIdx.x * blockDim.x + threadIdx.x;
    if (bf >= BF) return;
    float Lr[64], Li[64], Vr[64], Vi[64], Sr[64], Si[64];
    const float* pn = psd_n_g + (size_t)bf * 128;
    const float* ps = psd_s_g + (size_t)bf * 128;

    for (int i = 0; i < 64; ++i) { Lr[i] = pn[2 * i]; Li[i] = pn[2 * i + 1]; }
    float tr = 0.f;
    for (int i = 0; i < 8; ++i) tr += Lr[i * 8 + i];
    float reg = DIAG_REG * tr + EPS;
    for (int i = 0; i < 8; ++i) Lr[i * 8 + i] += reg;

    // in-place complex Cholesky (lower)
    for (int j = 0; j < 8; ++j) {
        float d = Lr[j * 8 + j];
        for (int k = 0; k < j; ++k)
            d -= Lr[j * 8 + k] * Lr[j * 8 + k] + Li[j * 8 + k] * Li[j * 8 + k];
        d = sqrtf(fmaxf(d, 1e-30f));
        Lr[j * 8 + j] = d; Li[j * 8 + j] = 0.f;
        float inv = 1.f / d;
        for (int i = j + 1; i < 8; ++i) {
            float sr = Lr[i * 8 + j], si = Li[i * 8 + j];
            for (int k = 0; k < j; ++k) {
                sr -= Lr[i * 8 + k] * Lr[j * 8 + k] + Li[i * 8 + k] * Li[j * 8 + k];
                si -= Li[i * 8 + k] * Lr[j * 8 + k] - Lr[i * 8 + k] * Li[j * 8 + k];
            }
            Lr[i * 8 + j] = sr * inv; Li[i * 8 + j] = si * inv;
        }
    }

    // invert lower triangular L -> V
    for (int i = 0; i < 64; ++i) { Vr[i] = 0.f; Vi[i] = 0.f; }
    for (int j = 0; j < 8; ++j) {
        Vr[j * 8 + j] = 1.f / Lr[j * 8 + j];
        for (int i = j + 1; i < 8; ++i) {
            float sr = 0.f, si = 0.f;
            for (int k = j; k < i; ++k) {
                sr += Lr[i * 8 + k] * Vr[k * 8 + j] - Li[i * 8 + k] * Vi[k * 8 + j];
                si += Lr[i * 8 + k] * Vi[k * 8 + j] + Li[i * 8 + k] * Vr[k * 8 + j];
            }
            float invd = 1.f / Lr[i * 8 + i];
            Vr[i * 8 + j] = -sr * invd;
            Vi[i * 8 + j] = -si * invd;
        }
    }

    // S <- V * psd_s * V^H  (in place; V lower triangular)
    for (int i = 0; i < 64; ++i) { Sr[i] = ps[2 * i]; Si[i] = ps[2 * i + 1]; }
    for (int i = 7; i >= 0; --i)            // M = V*S, bottom-up rows
        for (int col = 0; col < 8; ++col) {
            float ar = 0.f, ai = 0.f;
            for (int k = 0; k <= i; ++k) {
                ar += Vr[i * 8 + k] * Sr[k * 8 + col] - Vi[i * 8 + k] * Si[k * 8 + col];
                ai += Vr[i * 8 + k] * Si[k * 8 + col] + Vi[i * 8 + k] * Sr[k * 8 + col];
            }
            Sr[i * 8 + col] = ar; Si[i * 8 + col] = ai;
        }
    for (int j = 7; j >= 0; --j)            // W = M*V^H, descending cols
        for (int r = 0; r < 8; ++r) {
            float ar = 0.f, ai = 0.f;
            for (int k = 0; k <= j; ++k) {
                ar += Sr[r * 8 + k] * Vr[j * 8 + k] + Si[r * 8 + k] * Vi[j * 8 + k];
                ai += Si[r * 8 + k] * Vr[j * 8 + k] - Sr[r * 8 + k] * Vi[j * 8 + k];
            }
            Sr[r * 8 + j] = ar; Si[r * 8 + j] = ai;
        }

    // power iteration: principal eigenvector of Hermitian PSD matrix
    float vr[8], vi[8];
    for (int i = 0; i < 8; ++i) { vr[i] = 1.f; vi[i] = 0.f; }
    for (int it = 0; it < 50; ++it) {
        float wr[8], wi[8], nrm = 0.f;
        for (int r = 0; r < 8; ++r) {
            float ar = 0.f, ai = 0.f;
            for (int k = 0; k < 8; ++k) {
                ar += Sr[r * 8 + k] * vr[k] - Si[r * 8 + k] * vi[k];
                ai += Sr[r * 8 + k] * vi[k] + Si[r * 8 + k] * vr[k];
            }
            wr[r] = ar; wi[r] = ai; nrm += ar * ar + ai * ai;
        }
        float rn = rsqrtf(fmaxf(nrm, 1e-38f));
        for (int r = 0; r < 8; ++r) { vr[r] = wr[r] * rn; vi[r] = wi[r] * rn; }
    }

    // sv = L * v
    float* svp = sv_g + (size_t)bf * 16;
    for (int r = 0; r < 8; ++r) {
        float ar = 0.f, ai = 0.f;
        for (int k = 0; k <= r; ++k) {
            ar += Lr[r * 8 + k] * vr[k] - Li[r * 8 + k] * vi[k];
            ai += Lr[r * 8 + k] * vi[k] + Li[r * 8 + k] * vr[k];
        }
        svp[2 * r] = ar; svp[2 * r + 1] = ai;
    }
}

// ------------------------------------------------------- per-time weighting
__global__ void weight_kernel(const float* __restrict__ xr, const float* __restrict__ xi,
                              const float* __restrict__ outc, float* __restrict__ weight,
                              int first)
{
    long idx = (long)blockIdx.x * blockDim.x + threadIdx.x;
    if (idx >= (long)BF * TT) return;
    int bf = (int)(idx / TT);
    int t  = (int)(idx - (long)bf * TT);
    int b = bf / FF, f = bf - b * FF;
    float p;
    if (first) {
        p = 0.f;
        for (int c = 0; c < CC; ++c) {
            size_t o = (((size_t)b * CC + c) * FF + f) * (size_t)TT + t;
            float r = xr[o], im = xi[o];
            p += r * r + im * im;
        }
        p *= (1.0f / CC);
    } else {
        size_t o = ((size_t)bf * TT + t) * 2;
        float r = outc[o], im = outc[o + 1];
        p = r * r + im * im;
    }
    weight[idx] = 1.0f / fmaxf(p, EPS);
}

// ------------------------------------- MPDR solve: W = R^-1 h / (h^H R^-1 h)
__global__ void mpdr_solve_kernel(const float* __restrict__ psd_w_g,
                                  const float* __restrict__ sv_g,
                                  float* __restrict__ W_g)
{
    int bf = blockIdx.x * blockDim.x + threadIdx.x;
    if (bf >= BF) return;
    float Lr[64], Li[64];
    const float* pw = psd_w_g + (size_t)bf * 128;
    for (int i = 0; i < 64; ++i) { Lr[i] = pw[2 * i]; Li[i] = pw[2 * i + 1]; }
    float tr = 0.f;
    for (int i = 0; i < 8; ++i) tr += Lr[i * 8 + i];
    float reg = DIAG_REG * tr + EPS;
    for (int i = 0; i < 8; ++i) Lr[i * 8 + i] += reg;

    for (int j = 0; j < 8; ++j) {           // complex Cholesky
        float d = Lr[j * 8 + j];
        for (int k = 0; k < j; ++k)
            d -= Lr[j * 8 + k] * Lr[j * 8 + k] + Li[j * 8 + k] * Li[j * 8 + k];
        d = sqrtf(fmaxf(d, 1e-30f));
        Lr[j * 8 + j] = d; Li[j * 8 + j] = 0.f;
        float inv = 1.f / d;
        for (int i = j + 1; i < 8; ++i) {
            float sr = Lr[i * 8 + j], si = Li[i * 8 + j];
            for (int k = 0; k < j; ++k) {
                sr -= Lr[i * 8 + k] * Lr[j * 8 + k] + Li[i * 8 + k] * Li[j * 8 + k];
                si -= Li[i * 8 + k] * Lr[j * 8 + k] - Lr[i * 8 + k] * Li[j * 8 + k];
            }
            Lr[i * 8 + j] = sr * inv; Li[i * 8 + j] = si * inv;
        }
    }

    const float* svp = sv_g + (size_t)bf * 16;
    float hr[8], hi[8], yr[8], yi[8], zr[8], zi[8];
    for (int i = 0; i < 8; ++i) { hr[i] = svp[2 * i]; hi[i] = svp[2 * i + 1]; }

    for (int i = 0; i < 8; ++i) {           // L y = h
        float ar = hr[i], ai = hi[i];
        for (int k = 0; k < i; ++k) {
            ar -= Lr[i * 8 + k] * yr[k] - Li[i * 8 + k] * yi[k];
            ai -= Lr[i * 8 + k] * yi[k] + Li[i * 8 + k] * yr[k];
        }
        float inv = 1.f / Lr[i * 8 + i];
        yr[i] = ar * inv; yi[i] = ai * inv;
    }
    for (int i = 7; i >= 0; --i) {          // L^H z = y
        float ar = yr[i], ai = yi[i];
        for (int k = i + 1; k < 8; ++k) {
            ar -= Lr[k * 8 + i] * zr[k] + Li[k * 8 + i] * zi[k];
            ai -= Lr[k * 8 + i] * zi[k] - Li[k * 8 + i] * zr[k];
        }
        float inv = 1.f / Lr[i * 8 + i];
        zr[i] = ar * inv; zi[i] = ai * inv;
    }

    float dr = EPS, di = 0.f;               // denom = h^H z + eps
    for (int k = 0; k < 8; ++k) {
        dr += hr[k] * zr[k] + hi[k] * zi[k];
        di += hr[k] * zi[k] - hi[k] * zr[k];
    }
    float idn = 1.0f / fmaxf(dr * dr + di * di, 1e-38f);
    float* Wp = W_g + (size_t)bf * 16;
    for (int k = 0; k < 8; ++k) {           // W = z / denom
        Wp[2 * k]     = (zr[k] * dr + zi[k] * di) * idn;
        Wp[2 * k + 1] = (zi[k] * dr - zr[k] * di) * idn;
    }
}

// -------------------------------------------------- out = W^H x  (B,F,T cplx)
__global__ void apply_kernel(const float* __restrict__ xr, const float* __restrict__ xi,
                             const float* __restrict__ W_g, float* __restrict__ outc)
{
    long idx = (long)blockIdx.x * blockDim.x + threadIdx.x;
    if (idx >= (long)BF * TT) return;
    int bf = (int)(idx / TT);
    int t  = (int)(idx - (long)bf * TT);
    int b = bf / FF, f = bf - b * FF;
    const float* Wp = W_g + (size_t)bf * 16;
    float ar = 0.f, ai = 0.f;
    for (int c = 0; c < CC; ++c) {
        size_t o = (((size_t)b * CC + c) * FF + f) * (size_t)TT + t;
        float r = xr[o], im = xi[o];
        float wr = Wp[2 * c], wi = Wp[2 * c + 1];
        ar += wr * r + wi * im;             // conj(W)*x
        ai += wr * im - wi * r;
    }
    size_t o2 = ((size_t)bf * TT + t) * 2;  // (B,1,F,T,2) interleaved
    outc[o2] = ar; outc[o2 + 1] = ai;
}

extern "C" void kernel_launch(void* const* d_in, const int* in_sizes, int n_in,
                              void* d_out, int out_size, void* d_ws, size_t ws_size,
                              hipStream_t stream) {
    const float* xr   = (const float*)d_in[0];
    const float* xi   = (const float*)d_in[1];
    const float* mask = (const float*)d_in[2];
    float* out = (float*)d_out;

    float* wsf    = (float*)d_ws;
    float* msum   = wsf;                     // BF
    float* psd_s  = msum  + BF;              // BF*128
    float* psd_n  = psd_s + (size_t)BF * 128;
    float* psd_w  = psd_n + (size_t)BF * 128;
    float* sv     = psd_w + (size_t)BF * 128; // BF*16
    float* Wf     = sv    + (size_t)BF * 16;  // BF*16
    float* weight = Wf    + (size_t)BF * 16;  // BF*T

    const long nbt = (long)BF * TT;
    const int  thr = 256;
    const int  gBT = (int)((nbt + thr - 1) / thr);

    msum_kernel<<<(BF + thr - 1) / thr, thr, 0, stream>>>(mask, msum);
    psd_wmma_kernel<<<BF, 128, 0, stream>>>(xr, xi, mask, msum, psd_s, 0);
    psd_wmma_kernel<<<BF, 128, 0, stream>>>(xr, xi, mask, msum, psd_n, 1);
    steering_kernel<<<(BF + 63) / 64, 64, 0, stream>>>(psd_s, psd_n, sv);

    for (int it = 0; it < NUM_ITER; ++it) {
        weight_kernel<<<gBT, thr, 0, stream>>>(xr, xi, out, weight, it == 0 ? 1 : 0);
        psd_wmma_kernel<<<BF, 128, 0, stream>>>(xr, xi, weight, msum, psd_w, 2);
        mpdr_solve_kernel<<<(BF + 63) / 64, 64, 0, stream>>>(psd_w, sv, Wf);
        apply_kernel<<<gBT, thr, 0, stream>>>(xr, xi, Wf, out);
    }
}